// LossWasserstein_7627861918343
// MI455X (gfx1250) — compile-verified
//
#include <hip/hip_runtime.h>
#include <math.h>

// ---------------- problem constants (match reference) ----------------
#define NPT 2048
#define MPT 2048
#define KC  8
#define DD  64

typedef __attribute__((ext_vector_type(2))) float v2f;
typedef __attribute__((ext_vector_type(8))) float v8f;

// ---------------- workspace layout (bytes) ----------------
static constexpr size_t SZMAT   = (size_t)NPT * MPT * sizeof(float);   // 16 MB
static constexpr size_t OFF_CXY = 0;
static constexpr size_t OFF_CYX = SZMAT;
static constexpr size_t OFF_CXX = 2 * SZMAT;
static constexpr size_t OFF_CYY = 3 * SZMAT;
static constexpr size_t OFF_XP  = 4 * SZMAT;                 // permuted x   [N,64]
static constexpr size_t OFF_YP  = OFF_XP + (size_t)NPT*DD*4; // permuted y   [M,64]
static constexpr size_t OFF_SQX = OFF_YP + (size_t)MPT*DD*4; // |x|^2 permuted
static constexpr size_t OFF_SQY = OFF_SQX + NPT*4;
// potentials (packed, permuted order) -- FAB..GBT contiguous for clearing
static constexpr size_t OFF_FAB = OFF_SQY + MPT*4;
static constexpr size_t OFF_GAB = OFF_FAB + NPT*4;
static constexpr size_t OFF_FAA = OFF_GAB + MPT*4;
static constexpr size_t OFF_GBB = OFF_FAA + NPT*4;
static constexpr size_t OFF_FT  = OFF_GBB + MPT*4;
static constexpr size_t OFF_GT  = OFF_FT  + NPT*4;
static constexpr size_t OFF_FAT = OFF_GT  + MPT*4;
static constexpr size_t OFF_GBT = OFF_FAT + NPT*4;
static constexpr size_t OFF_PREDX = OFF_GBT + MPT*4;
static constexpr size_t OFF_RANKX = OFF_PREDX + NPT*4;
static constexpr size_t OFF_RANKY = OFF_RANKX + NPT*4;
static constexpr size_t OFF_CLUSX = OFF_RANKY + MPT*4;
static constexpr size_t OFF_CLUSY = OFF_CLUSX + NPT*4;
static constexpr size_t OFF_CNTX  = OFF_CLUSY + MPT*4;  // int[8] ; CNTY must follow
static constexpr size_t OFF_CNTY  = OFF_CNTX + 32;
static constexpr size_t OFF_SX    = OFF_CNTY + 32;
static constexpr size_t OFF_SY    = OFF_SX + 32;
static constexpr size_t OFF_LWX   = OFF_SY + 32;
static constexpr size_t OFF_LWY   = OFF_LWX + 32;
static constexpr size_t OFF_VALID = OFF_LWY + 32;
static constexpr size_t OFF_ICX   = OFF_VALID + 32;
static constexpr size_t OFF_ICY   = OFF_ICX + 32;
static constexpr size_t OFF_LFIL  = OFF_ICY + 32;

#define WSF(off) ((float*)(ws + (off)))
#define WSI(off) ((int*)(ws + (off)))

// ---------------- clear counters + potentials ----------------
__global__ void k_clear(char* ws) {
    int tid = blockIdx.x * blockDim.x + threadIdx.x;
    float* pots = WSF(OFF_FAB);                    // 8 arrays x 2048 f32, contiguous
    for (int i = tid; i < 8 * NPT; i += gridDim.x * blockDim.x) pots[i] = 0.f;
    if (tid < 16) WSI(OFF_CNTX)[tid] = 0;          // cntx[8] ++ cnty[8]
}

// ---------------- kmeans assign + histogram ----------------
__global__ void k_assign(const float* __restrict__ x, const float* __restrict__ cc,
                         const int* __restrict__ predy, char* ws) {
    int tid = blockIdx.x * blockDim.x + threadIdx.x;
    if (tid < NPT) {
        const float* xi = x + (size_t)tid * DD;
        float best = 3.4e38f; int bk = 0;
        for (int k = 0; k < KC; ++k) {
            const float* ck = cc + k * DD;
            float d = 0.f;
            for (int dd = 0; dd < DD; ++dd) { float df = xi[dd] - ck[dd]; d = fmaf(df, df, d); }
            if (d < best) { best = d; bk = k; }    // first-min tie-break (argmin)
        }
        WSI(OFF_PREDX)[tid] = bk;
        atomicAdd(WSI(OFF_CNTX) + bk, 1);
    } else if (tid < NPT + MPT) {
        int j = tid - NPT;
        atomicAdd(WSI(OFF_CNTY) + predy[j], 1);
    }
}

// ---------------- deterministic stable rank within cluster ----------------
__global__ void k_rank(const int* __restrict__ predy, char* ws) {
    int tid = blockIdx.x * blockDim.x + threadIdx.x;
    if (tid < NPT) {
        const int* px = WSI(OFF_PREDX);
        int k = px[tid], r = 0;
        for (int j = 0; j < tid; ++j) r += (px[j] == k);
        WSI(OFF_RANKX)[tid] = r;
    } else if (tid < NPT + MPT) {
        int j = tid - NPT, k = predy[j], r = 0;
        for (int jj = 0; jj < j; ++jj) r += (predy[jj] == k);
        WSI(OFF_RANKY)[j] = r;
    }
}

// ---------------- segment starts, log-weights, validity, filling loss ----------------
__global__ void k_meta(const float* __restrict__ fill, char* ws) {
    if (blockIdx.x || threadIdx.x) return;
    int sx = 0, sy = 0; float lf = 0.f;
    for (int k = 0; k < KC; ++k) {
        int cx = WSI(OFF_CNTX)[k], cy = WSI(OFF_CNTY)[k];
        WSI(OFF_SX)[k] = sx; WSI(OFF_SY)[k] = sy; sx += cx; sy += cy;
        WSF(OFF_LWX)[k] = -__logf(fmaxf((float)cx, 1.f));
        WSF(OFF_LWY)[k] = -__logf(fmaxf((float)cy, 1.f));
        WSI(OFF_VALID)[k] = (cx > 0 && cy > 0) ? 1 : 0;
        WSF(OFF_ICX)[k] = 1.f / fmaxf((float)cx, 1.f);
        WSF(OFF_ICY)[k] = 1.f / fmaxf((float)cy, 1.f);
        float df = (float)cx / (float)NPT - fill[k];
        lf = fmaf(df, df, lf);
    }
    WSF(OFF_LFIL)[0] = lf / (float)KC;
}

// ---------------- cluster-sorted scatter of points + squared norms ----------------
__global__ void k_scatter(const float* __restrict__ x, const float* __restrict__ y,
                          const int* __restrict__ predy, char* ws) {
    int tid = blockIdx.x * blockDim.x + threadIdx.x;
    if (tid < NPT) {
        int k = WSI(OFF_PREDX)[tid];
        int pos = WSI(OFF_SX)[k] + WSI(OFF_RANKX)[tid];
        float s = 0.f;
        for (int dd = 0; dd < DD; ++dd) {
            float v = x[(size_t)tid * DD + dd];
            WSF(OFF_XP)[(size_t)pos * DD + dd] = v;
            s = fmaf(v, v, s);
        }
        WSF(OFF_SQX)[pos] = s; WSI(OFF_CLUSX)[pos] = k;
    } else if (tid < NPT + MPT) {
        int j = tid - NPT, k = predy[j];
        int pos = WSI(OFF_SY)[k] + WSI(OFF_RANKY)[j];
        float s = 0.f;
        for (int dd = 0; dd < DD; ++dd) {
            float v = y[(size_t)j * DD + dd];
            WSF(OFF_YP)[(size_t)pos * DD + dd] = v;
            s = fmaf(v, v, s);
        }
        WSF(OFF_SQY)[pos] = s; WSI(OFF_CLUSY)[pos] = k;
    }
}

// ---------------- fp32 WMMA cost GEMM: C = 0.5*max(|a|^2+|b|^2-2 a.b, 0) ----------------
// One wave -> one 16x16 tile. K=64 in 16 chunks of V_WMMA_F32_16X16X4_F32.
// A 16x4 f32 layout: lanes 0-15 M=lane {K=k0,k0+1}; lanes 16-31 M=lane-16 {K=k0+2,k0+3}.
// B 4x16 mirrors (rows striped across lanes within a VGPR). C/D: v8f, lanes 0-15 M=v, 16-31 M=v+8.
__global__ void __launch_bounds__(256) k_cost(const float* __restrict__ A, const float* __restrict__ B,
                                              const float* __restrict__ sqA, const float* __restrict__ sqB,
                                              float* __restrict__ Cout) {
    int lane = threadIdx.x & 31;
    int wave = (blockIdx.x * blockDim.x + threadIdx.x) >> 5;
    const int tilesN = MPT / 16;                       // 128
    int tm = (wave / tilesN) * 16, tn = (wave % tilesN) * 16;
    int half = lane >> 4, l = lane & 15;

    v8f acc = {0.f, 0.f, 0.f, 0.f, 0.f, 0.f, 0.f, 0.f};
    const float* arow = A + (size_t)(tm + l) * DD + 2 * half;
    const float* brow = B + (size_t)(tn + l) * DD + 2 * half;
#pragma unroll
    for (int k0 = 0; k0 < DD; k0 += 4) {
        v2f a; a.x = arow[k0]; a.y = arow[k0 + 1];
        v2f b; b.x = brow[k0]; b.y = brow[k0 + 1];
        acc = __builtin_amdgcn_wmma_f32_16x16x4_f32(false, a, false, b, (short)0, acc, false, false);
    }
    int col = tn + l;
    float sb = sqB[col];
#pragma unroll
    for (int v = 0; v < 8; ++v) {
        int row = tm + v + 8 * half;
        float val = 0.5f * fmaxf(sqA[row] + sb - 2.f * acc[v], 0.f);
        Cout[(size_t)row * MPT + col] = val;
    }
}

// ---------------- block-diagonal softmin (all 4 directions in one launch) ----------------
// Row r in [0,4N): task = r>>11. One wave per row; two passes (max, exp-sum) over the
// row's own-cluster column segment (other clusters are exactly zero through -1e9 mask).
__global__ void __launch_bounds__(256) k_softmin(char* ws, float eps) {
    int wave = threadIdx.x >> 5, lane = threadIdx.x & 31;
    int r = blockIdx.x * 8 + wave;
    int task = r >> 11, i = r & (NPT - 1);

    const float *Cm, *pot, *lw; const int *seg, *cnt, *clus; float* out;
    switch (task) {
        case 0:  Cm = WSF(OFF_CXY); pot = WSF(OFF_GAB); lw = WSF(OFF_LWY);
                 seg = WSI(OFF_SY); cnt = WSI(OFF_CNTY); clus = WSI(OFF_CLUSX); out = WSF(OFF_FT);  break;
        case 1:  Cm = WSF(OFF_CYX); pot = WSF(OFF_FAB); lw = WSF(OFF_LWX);
                 seg = WSI(OFF_SX); cnt = WSI(OFF_CNTX); clus = WSI(OFF_CLUSY); out = WSF(OFF_GT);  break;
        case 2:  Cm = WSF(OFF_CXX); pot = WSF(OFF_FAA); lw = WSF(OFF_LWX);
                 seg = WSI(OFF_SX); cnt = WSI(OFF_CNTX); clus = WSI(OFF_CLUSX); out = WSF(OFF_FAT); break;
        default: Cm = WSF(OFF_CYY); pot = WSF(OFF_GBB); lw = WSF(OFF_LWY);
                 seg = WSI(OFF_SY); cnt = WSI(OFF_CNTY); clus = WSI(OFF_CLUSY); out = WSF(OFF_GBT); break;
    }
    int k = clus[i], s0 = seg[k], len = cnt[k];
    if (len == 0) { if (lane == 0) out[i] = 0.f; return; }   // invalid cluster: inert

    const float* crow = Cm + (size_t)i * MPT + s0;
    const float* pp = pot + s0;
    float inv = 1.f / eps;

    float m = -3.4e38f;
    for (int j = lane; j < len; j += 32) m = fmaxf(m, (pp[j] - crow[j]) * inv);
    for (int off = 16; off; off >>= 1) m = fmaxf(m, __shfl_xor(m, off, 32));

    float s = 0.f;
    for (int j = lane; j < len; j += 32) s += __expf((pp[j] - crow[j]) * inv - m);
    for (int off = 16; off; off >>= 1) s += __shfl_xor(s, off, 32);

    if (lane == 0) out[i] = -eps * (lw[k] + m + __logf(s));
}

// ---------------- symmetric averaged update ----------------
__global__ void k_combine(char* ws) {
    int i = blockIdx.x * blockDim.x + threadIdx.x;   // 2048 threads
    WSF(OFF_FAB)[i] = 0.5f * (WSF(OFF_FAB)[i] + WSF(OFF_FT)[i]);
    WSF(OFF_GAB)[i] = 0.5f * (WSF(OFF_GAB)[i] + WSF(OFF_GT)[i]);
    WSF(OFF_FAA)[i] = 0.5f * (WSF(OFF_FAA)[i] + WSF(OFF_FAT)[i]);
    WSF(OFF_GBB)[i] = 0.5f * (WSF(OFF_GBB)[i] + WSF(OFF_GBT)[i]);
}

// ---------------- final reduction: S = <a, f-faa> + <b, g-gbb> (+ filling mse) ----------
__global__ void k_final(char* ws, float* __restrict__ out) {
    __shared__ float red[256];
    int tid = threadIdx.x;
    float acc = 0.f;
    for (int i = tid; i < NPT; i += 256) {
        int kx = WSI(OFF_CLUSX)[i];
        if (WSI(OFF_VALID)[kx]) acc += (WSF(OFF_FT)[i] - WSF(OFF_FAT)[i]) * WSF(OFF_ICX)[kx];
        int ky = WSI(OFF_CLUSY)[i];
        if (WSI(OFF_VALID)[ky]) acc += (WSF(OFF_GT)[i] - WSF(OFF_GBT)[i]) * WSF(OFF_ICY)[ky];
    }
    red[tid] = acc; __syncthreads();
    for (int o = 128; o; o >>= 1) { if (tid < o) red[tid] += red[tid + o]; __syncthreads(); }
    if (tid == 0) out[0] = red[0] + WSF(OFF_LFIL)[0];
}

// ---------------- host driver ----------------
extern "C" void kernel_launch(void* const* d_in, const int* in_sizes, int n_in,
                              void* d_out, int out_size, void* d_ws, size_t ws_size,
                              hipStream_t stream) {
    (void)in_sizes; (void)n_in; (void)out_size; (void)ws_size;
    const float* x     = (const float*)d_in[0];
    const float* cc    = (const float*)d_in[1];
    const float* fillt = (const float*)d_in[2];
    const float* tgt   = (const float*)d_in[3];
    const int*   predy = (const int*)d_in[4];
    char* ws = (char*)d_ws;
    float* out = (float*)d_out;

    k_clear  <<<64, 256, 0, stream>>>(ws);
    k_assign <<<(NPT + MPT) / 256, 256, 0, stream>>>(x, cc, predy, ws);
    k_rank   <<<(NPT + MPT) / 256, 256, 0, stream>>>(predy, ws);
    k_meta   <<<1, 1, 0, stream>>>(fillt, ws);
    k_scatter<<<(NPT + MPT) / 256, 256, 0, stream>>>(x, tgt, predy, ws);

    // 4 Gram/cost matrices via fp32 WMMA (16384 waves each -> 2048 blocks of 8 waves)
    float* Xp = WSF(OFF_XP); float* Yp = WSF(OFF_YP);
    float* sx = WSF(OFF_SQX); float* sy = WSF(OFF_SQY);
    k_cost<<<2048, 256, 0, stream>>>(Xp, Yp, sx, sy, WSF(OFF_CXY));
    k_cost<<<2048, 256, 0, stream>>>(Yp, Xp, sy, sx, WSF(OFF_CYX));
    k_cost<<<2048, 256, 0, stream>>>(Xp, Xp, sx, sx, WSF(OFF_CXX));
    k_cost<<<2048, 256, 0, stream>>>(Yp, Yp, sy, sy, WSF(OFF_CYY));

    // eps-scaling schedule: e = diam^2 * (scaling^2)^t down to blur^2 (geomloss p=2)
    double e = 16.0, emin = 0.05 * 0.05, ratio = 0.8 * 0.8;
    float epslist[64]; int ne = 0;
    while (e > emin && ne < 62) { epslist[ne++] = (float)e; e *= ratio; }
    epslist[ne++] = (float)emin;                       // 21 steps

    for (int t = 0; t < ne; ++t) {
        k_softmin<<<(4 * NPT) / 8, 256, 0, stream>>>(ws, epslist[t]);
        k_combine<<<NPT / 256, 256, 0, stream>>>(ws);
    }
    // final extrapolation at eps_min (no averaging), then reduce
    k_softmin<<<(4 * NPT) / 8, 256, 0, stream>>>(ws, epslist[ne - 1]);
    k_final  <<<1, 256, 0, stream>>>(ws, out);
}